// MusicalRNN_87333864997690
// MI455X (gfx1250) — compile-verified
//
#include <hip/hip_runtime.h>

typedef float v2f __attribute__((ext_vector_type(2)));
typedef float v8f __attribute__((ext_vector_type(8)));

#define T_SEQ   2048
#define N_BATCH 512
#define N_HID   20
#define N_OUT   255
#define N_LAYER 3

// ---------- helpers ------------------------------------------------------

static __device__ __forceinline__ float shflf(float v, int srcLane) {
    return __int_as_float(__builtin_amdgcn_ds_bpermute(srcLane << 2, __float_as_int(v)));
}

static __device__ __forceinline__ float fast_tanh(float x) {
    // tanh(x) = (e^{2x}-1)/(e^{2x}+1), e^{2x} = 2^{x*2*log2(e)}; clamp to avoid inf/inf
    x = fminf(15.0f, fmaxf(-15.0f, x));
    float e = __builtin_amdgcn_exp2f(x * 2.885390081777927f);
    return (e - 1.0f) * __builtin_amdgcn_rcpf(e + 1.0f);
}

static __device__ __forceinline__ v8f wmma4(v2f a, v2f b, v8f c) {
    // D = A(16x4 f32) * B(4x16 f32) + C(16x16 f32)
    return __builtin_amdgcn_wmma_f32_16x16x4_f32(
        /*neg_a=*/false, a, /*neg_b=*/false, b,
        /*c_mod=*/(short)0, c, /*reuse_a=*/false, /*reuse_b=*/false);
}

// Load one A-matrix fragment (16x4) of a HIDxHID row-major weight matrix.
// A layout: lanes0-15 hold row M=lane (K = 4k..4k+1), lanes16-31 row M=lane-16 (K = 4k+2..4k+3).
static __device__ __forceinline__ v2f load_w_afrag(const float* __restrict__ W,
                                                   int mt, int k, int lane) {
    int half = lane >> 4;
    int m  = mt * 16 + (lane & 15);
    int kk = k * 4 + half * 2;
    v2f r;
    if (m < N_HID) { r.x = W[m * N_HID + kk]; r.y = W[m * N_HID + kk + 1]; }
    else           { r.x = 0.0f;              r.y = 0.0f; }
    return r;
}

// Convert tanh'ed C/D-layout (t0: hid 0..15, t1: hid 16..19) into B-layout frags
// g[k]: comp.x = (hid 4k   | 4k+2 in high half), comp.y = (hid 4k+1 | 4k+3), batch = lane&15.
static __device__ __forceinline__ void cvt_to_b(const float t0[8], const float t1[4],
                                                v2f g[5], int lane) {
    const int  lo  = lane & 15;
    const bool low = lane < 16;
    { float s0 = shflf(t0[2], lo);      float s1 = shflf(t0[3], lo);
      g[0].x = low ? t0[0] : s0;        g[0].y = low ? t0[1] : s1; }
    { float s0 = shflf(t0[6], lo);      float s1 = shflf(t0[7], lo);
      g[1].x = low ? t0[4] : s0;        g[1].y = low ? t0[5] : s1; }
    { float s0 = shflf(t0[0], lo + 16); float s1 = shflf(t0[1], lo + 16);
      g[2].x = low ? s0 : t0[2];        g[2].y = low ? s1 : t0[3]; }
    { float s0 = shflf(t0[4], lo + 16); float s1 = shflf(t0[5], lo + 16);
      g[3].x = low ? s0 : t0[6];        g[3].y = low ? s1 : t0[7]; }
    { float s0 = shflf(t1[2], lo);      float s1 = shflf(t1[3], lo);
      g[4].x = low ? t1[0] : s0;        g[4].y = low ? t1[1] : s1; }
}

// ---------- kernel 1: the recurrence (32 waves, one 16-batch tile each) ----

__global__ void __launch_bounds__(32)
rnn_kernel(const float* __restrict__ x,    const float* __restrict__ h0in,
           const float* __restrict__ Wih0, const float* __restrict__ Whh0,
           const float* __restrict__ bih0, const float* __restrict__ bhh0,
           const float* __restrict__ Wih1, const float* __restrict__ Whh1,
           const float* __restrict__ bih1, const float* __restrict__ bhh1,
           const float* __restrict__ Wih2, const float* __restrict__ Whh2,
           const float* __restrict__ bih2, const float* __restrict__ bhh2,
           float* __restrict__ hid_out, float* __restrict__ h2last)
{
    const int lane  = (int)threadIdx.x;
    const int lo    = lane & 15;
    const int half  = lane >> 4;
    const int bt    = (int)blockIdx.x;     // batch tile 0..31
    const int bbase = bt * 16;

    // Resident weight A-fragments: 5 matrices x 2 M-tiles x 5 K-frags
    v2f aWhh[3][2][5], aWih[2][2][5];
    #pragma unroll
    for (int mt = 0; mt < 2; ++mt)
        #pragma unroll
        for (int k = 0; k < 5; ++k) {
            aWhh[0][mt][k] = load_w_afrag(Whh0, mt, k, lane);
            aWhh[1][mt][k] = load_w_afrag(Whh1, mt, k, lane);
            aWhh[2][mt][k] = load_w_afrag(Whh2, mt, k, lane);
            aWih[0][mt][k] = load_w_afrag(Wih1, mt, k, lane);
            aWih[1][mt][k] = load_w_afrag(Wih2, mt, k, lane);
        }

    // Combined biases in C/D layout (per lane) + layer-0 input weight column
    const float* bi[3] = {bih0, bih1, bih2};
    const float* bh[3] = {bhh0, bhh1, bhh2};
    float pre_t0[3][8], pre_t1[3][4];
    #pragma unroll
    for (int l = 0; l < 3; ++l) {
        #pragma unroll
        for (int r = 0; r < 8; ++r) { int m = r + 8 * half; pre_t0[l][r] = bi[l][m] + bh[l][m]; }
        #pragma unroll
        for (int r = 0; r < 4; ++r) { pre_t1[l][r] = half ? 0.0f : (bi[l][16 + r] + bh[l][16 + r]); }
    }
    float w0_t0[8], w0_t1[4];
    #pragma unroll
    for (int r = 0; r < 8; ++r) w0_t0[r] = Wih0[r + 8 * half];
    #pragma unroll
    for (int r = 0; r < 4; ++r) w0_t1[r] = half ? 0.0f : Wih0[16 + r];

    // Hidden state in B-layout (G = H^T), from the "hidden" input
    v2f g[3][5];
    #pragma unroll
    for (int l = 0; l < 3; ++l)
        #pragma unroll
        for (int k = 0; k < 5; ++k) {
            int hidx = 4 * k + 2 * half;
            const float* hp = h0in + l * N_BATCH * N_HID + (bbase + lo) * N_HID + hidx;
            g[l][k].x = hp[0]; g[l][k].y = hp[1];
        }

    const bool rec_lane = (bt == 31) && (lo == 15);   // owns global batch row 511

    // Software-pipelined x stream: the load for step t is issued during step
    // t-1, so the only per-step memory access is off the recurrence critical
    // path (previously each iteration stalled on s_wait_loadcnt at the top).
    float xv = x[bbase + lo];                          // t = 0 value

    for (int t = 0; t < T_SEQ; ++t) {
        const int tn = (t + 1 < T_SEQ) ? (t + 1) : (T_SEQ - 1);
        const float xv_next = x[tn * N_BATCH + bbase + lo];   // prefetch t+1

        float t0[8], t1[4];

        // ---- layer 0: G0 = tanh(Wih0*x^T + Whh0*G0 + b)
        {
            v8f a0, a1;
            #pragma unroll
            for (int r = 0; r < 8; ++r) a0[r] = fmaf(w0_t0[r], xv, pre_t0[0][r]);
            #pragma unroll
            for (int r = 0; r < 4; ++r) a1[r] = fmaf(w0_t1[r], xv, pre_t1[0][r]);
            a1[4] = a1[5] = a1[6] = a1[7] = 0.0f;
            #pragma unroll
            for (int k = 0; k < 5; ++k) {
                a0 = wmma4(aWhh[0][0][k], g[0][k], a0);
                a1 = wmma4(aWhh[0][1][k], g[0][k], a1);
            }
            #pragma unroll
            for (int r = 0; r < 8; ++r) t0[r] = fast_tanh(a0[r]);
            #pragma unroll
            for (int r = 0; r < 4; ++r) t1[r] = fast_tanh(a1[r]);
            cvt_to_b(t0, t1, g[0], lane);
        }

        // ---- layers 1,2: Gl = tanh(Wih*G_{l-1} + Whh*Gl + b)
        #pragma unroll
        for (int l = 1; l < 3; ++l) {
            v8f a0, a1;
            #pragma unroll
            for (int r = 0; r < 8; ++r) a0[r] = pre_t0[l][r];
            #pragma unroll
            for (int r = 0; r < 4; ++r) a1[r] = pre_t1[l][r];
            a1[4] = a1[5] = a1[6] = a1[7] = 0.0f;
            #pragma unroll
            for (int k = 0; k < 5; ++k) {
                a0 = wmma4(aWih[l - 1][0][k], g[l - 1][k], a0);
                a1 = wmma4(aWih[l - 1][1][k], g[l - 1][k], a1);
                a0 = wmma4(aWhh[l][0][k],     g[l][k],     a0);
                a1 = wmma4(aWhh[l][1][k],     g[l][k],     a1);
            }
            #pragma unroll
            for (int r = 0; r < 8; ++r) t0[r] = fast_tanh(a0[r]);
            #pragma unroll
            for (int r = 0; r < 4; ++r) t1[r] = fast_tanh(a1[r]);
            cvt_to_b(t0, t1, g[l], lane);
        }

        // record relu(h2[t, 511, :]) — lanes 15 and 31 of the last tile hold it
        if (rec_lane) {
            #pragma unroll
            for (int k = 0; k < 5; ++k) {
                int hb = 4 * k + 2 * half;
                h2last[t * N_HID + hb]     = fmaxf(g[2][k].x, 0.0f);
                h2last[t * N_HID + hb + 1] = fmaxf(g[2][k].y, 0.0f);
            }
        }

        xv = xv_next;
    }

    // final hidden states (pre-relu) -> d_out hidden region
    #pragma unroll
    for (int l = 0; l < 3; ++l)
        #pragma unroll
        for (int k = 0; k < 5; ++k) {
            int hidx = 4 * k + 2 * half;
            float* hp = hid_out + l * N_BATCH * N_HID + (bbase + lo) * N_HID + hidx;
            hp[0] = g[l][k].x; hp[1] = g[l][k].y;
        }
}

// ---------- kernel 2: FC  logits[t][o] = relu_h2[t] . Wfc[o] + bfc[o] ------

__global__ void __launch_bounds__(32)
fc_kernel(const float* __restrict__ hseq, const float* __restrict__ Wfc,
          const float* __restrict__ bfc, float* __restrict__ logits)
{
    const int lane = (int)threadIdx.x;
    const int lo   = lane & 15;
    const int half = lane >> 4;
    const int mt   = (int)blockIdx.x;   // 0..127  (t tiles)
    const int nt   = (int)blockIdx.y;   // 0..15   (o tiles)
    const int o    = nt * 16 + lo;
    const bool ook = o < N_OUT;

    v2f af[5], bf[5];
    const int trow = mt * 16 + lo;
    #pragma unroll
    for (int k = 0; k < 5; ++k) {
        int kk = 4 * k + 2 * half;
        af[k].x = hseq[trow * N_HID + kk];
        af[k].y = hseq[trow * N_HID + kk + 1];
        if (ook) { bf[k].x = Wfc[o * N_HID + kk]; bf[k].y = Wfc[o * N_HID + kk + 1]; }
        else     { bf[k].x = 0.0f;                bf[k].y = 0.0f; }
    }
    float bv = ook ? bfc[o] : 0.0f;
    v8f acc;
    #pragma unroll
    for (int i = 0; i < 8; ++i) acc[i] = bv;
    #pragma unroll
    for (int k = 0; k < 5; ++k) acc = wmma4(af[k], bf[k], acc);

    if (ook) {
        #pragma unroll
        for (int r = 0; r < 8; ++r) {
            int tr = mt * 16 + r + 8 * half;
            logits[tr * N_OUT + o] = acc[r];
        }
    }
}

// ---------- launch --------------------------------------------------------

extern "C" void kernel_launch(void* const* d_in, const int* in_sizes, int n_in,
                              void* d_out, int out_size, void* d_ws, size_t ws_size,
                              hipStream_t stream) {
    (void)in_sizes; (void)n_in; (void)out_size; (void)ws_size;
    const float* x    = (const float*)d_in[0];
    const float* hid  = (const float*)d_in[1];
    const float* Wih0 = (const float*)d_in[2];
    const float* Whh0 = (const float*)d_in[3];
    const float* bih0 = (const float*)d_in[4];
    const float* bhh0 = (const float*)d_in[5];
    const float* Wih1 = (const float*)d_in[6];
    const float* Whh1 = (const float*)d_in[7];
    const float* bih1 = (const float*)d_in[8];
    const float* bhh1 = (const float*)d_in[9];
    const float* Wih2 = (const float*)d_in[10];
    const float* Whh2 = (const float*)d_in[11];
    const float* bih2 = (const float*)d_in[12];
    const float* bhh2 = (const float*)d_in[13];
    const float* Wfc  = (const float*)d_in[14];
    const float* bfc  = (const float*)d_in[15];

    float* logits  = (float*)d_out;                       // SEQ*OUT
    float* hid_out = (float*)d_out + T_SEQ * N_OUT;       // LAYERS*BATCH*HID
    float* hseq    = (float*)d_ws;                        // SEQ*HID scratch

    rnn_kernel<<<32, 32, 0, stream>>>(x, hid,
                                      Wih0, Whh0, bih0, bhh0,
                                      Wih1, Whh1, bih1, bhh1,
                                      Wih2, Whh2, bih2, bhh2,
                                      hid_out, hseq);
    fc_kernel<<<dim3(128, 16), 32, 0, stream>>>(hseq, Wfc, bfc, logits);
}